// NKATModel_18064632447058
// MI455X (gfx1250) — compile-verified
//
#include <hip/hip_runtime.h>
#include <hip/hip_bf16.h>

typedef float v2f __attribute__((ext_vector_type(2)));
typedef float v4f __attribute__((ext_vector_type(4)));
typedef float v8f __attribute__((ext_vector_type(8)));

#define B_TOT   65536
#define MM      16
#define IJ      64          // L*L
#define DIMX    4
#define WAVES   8
#define THREADS 256
#define BPW     4           // batches per wave (16 GEMM rows = 4 batches x 4 mu)

__device__ __forceinline__ float fast_rcpf(float v) {
#if __has_builtin(__builtin_amdgcn_rcpf)
    return __builtin_amdgcn_rcpf(v);
#else
    return 1.0f / v;
#endif
}

__device__ __forceinline__ float fast_tanhf(float v) {
#if __has_builtin(__builtin_amdgcn_tanhf)
    return __builtin_amdgcn_tanhf(v);
#else
    float e = __expf(2.0f * v);
    return 1.0f - 2.0f * fast_rcpf(e + 1.0f);
#endif
}

__global__ __launch_bounds__(THREADS)
void nkat_wmma_kernel(const float* __restrict__ x,
                      const float* __restrict__ a_, const float* __restrict__ b_,
                      const float* __restrict__ c_, const float* __restrict__ d_,
                      const float* __restrict__ alpha, const float* __restrict__ beta,
                      const float* __restrict__ gamma_, const float* __restrict__ delta,
                      const float* __restrict__ omega, const float* __restrict__ Lam,
                      float* __restrict__ out)
{
    // params transposed to [m][n] for conflict-free per-lane (n) access
    __shared__ float sAl[256], sBe[256], sGa[256], sDe[256], sOm[256];
    __shared__ v4f   sABCD[16];                       // {a,b,c,d} per n
    __shared__ __align__(16) v2f sS[WAVES * 64];      // per wave: [bl(4)][n(16)] -> {S1,S2}
    __shared__ v4f   sXv[WAVES * BPW];                // per wave: x[batch] (4 comps)

    const int tid  = threadIdx.x;
    const int lane = tid & 31;
    const int wave = tid >> 5;
    const int h    = lane >> 4;       // lane half (0/1)
    const int col  = lane & 15;       // N index within 16-wide tile

    // ---- stage scalar params into LDS (transposed) ----
    {
        int n = tid >> 4, m = tid & 15;
        int dst = m * 16 + n;
        sAl[dst] = alpha[tid];
        sBe[dst] = beta[tid];
        sGa[dst] = gamma_[tid];
        sDe[dst] = delta[tid];
        sOm[dst] = omega[tid];
        if (tid < 16) {
            v4f t = {a_[tid], b_[tid], c_[tid], d_[tid]};
            sABCD[tid] = t;
        }
    }

    // ---- load Lam as WMMA B-fragments: B[t][k], K = 4k + 2h + j, N = col ----
    // (constant for the whole kernel; 16 v2f = 32 VGPRs, reused by every WMMA)
    v2f Bf[4][4];
    #pragma unroll
    for (int t = 0; t < 4; ++t) {
        #pragma unroll
        for (int k = 0; k < 4; ++k) {
            int n0 = 4 * k + 2 * h;
            Bf[t][k].x = Lam[(size_t)n0       * IJ + t * 16 + col];
            Bf[t][k].y = Lam[(size_t)(n0 + 1) * IJ + t * 16 + col];
        }
    }

    __syncthreads();

    const int tile      = blockIdx.x * WAVES + wave;   // 0..16383
    const int baseBatch = tile * BPW;

    // ---- phase 1: S1/S2 for (bl, n); lane covers n = lane&15, bl = lane>>4 and bl+2 ----
    {
        int n   = lane & 15;
        int bl0 = lane >> 4;
        v4f x0 = *(const v4f*)(x + (size_t)(baseBatch + bl0)     * DIMX);
        v4f x1 = *(const v4f*)(x + (size_t)(baseBatch + bl0 + 2) * DIMX);
        float x2_0 = x0.x*x0.x + x0.y*x0.y + x0.z*x0.z + x0.w*x0.w;
        float x2_1 = x1.x*x1.x + x1.y*x1.y + x1.z*x1.z + x1.w*x1.w;
        float r0 = __fsqrt_rn(x2_0 + 1e-10f);
        float r1 = __fsqrt_rn(x2_1 + 1e-10f);

        float S1_0 = 0.f, S2_0 = 0.f, S1_1 = 0.f, S2_1 = 0.f;
        #pragma unroll
        for (int m = 0; m < MM; ++m) {
            int p = m * 16 + n;
            float al = sAl[p], be = sBe[p], ga = sGa[p], de = sDe[p], om = sOm[p];
            S1_0 += al * __expf(-be * x2_0);
            S1_1 += al * __expf(-be * x2_1);
            S2_0 += ga * __expf(-de * x2_0) * __cosf(om * r0);
            S2_1 += ga * __expf(-de * x2_1) * __cosf(om * r1);
        }
        v2f s0 = {S1_0, S2_0};
        v2f s1 = {S1_1, S2_1};
        sS[wave * 64 + bl0       * 16 + n] = s0;
        sS[wave * 64 + (bl0 + 2) * 16 + n] = s1;
        if (n == 0) {                     // one lane per bl publishes x
            sXv[wave * BPW + bl0]     = x0;
            sXv[wave * BPW + bl0 + 2] = x1;
        }
    }

    __syncthreads();

    // ---- phase 2: phi -> A fragments -> WMMA -> store ----
    {
        const int row = lane & 15;        // GEMM row within tile = bl*4 + mu
        const int bl  = row >> 2;
        const int mu  = row & 3;
        v4f xb = sXv[wave * BPW + bl];
        float xv = (mu == 0) ? xb.x : (mu == 1) ? xb.y : (mu == 2) ? xb.z : xb.w;

        v2f A[4];
        #pragma unroll
        for (int k = 0; k < 4; ++k) {
            int n0 = 4 * k + 2 * h;
            v4f s = *(const v4f*)&sS[wave * 64 + bl * 16 + n0];  // {S1[n0],S2[n0],S1[n0+1],S2[n0+1]}
            #pragma unroll
            for (int j = 0; j < 2; ++j) {
                float S1 = (j == 0) ? s.x : s.z;
                float S2 = (j == 0) ? s.y : s.w;
                v4f co = sABCD[n0 + j];
                float z  = S1 + S2 * xv;
                float t1 = fast_tanhf(co.x * z + co.y);
                float t2 = fast_tanhf(co.w * z);
                float ph = t1 + co.z * z * z * t2;
                if (j == 0) A[k].x = ph; else A[k].y = ph;
            }
        }

        v8f acc[4];
        #pragma unroll
        for (int t = 0; t < 4; ++t) {
            v8f z8 = {0.f, 0.f, 0.f, 0.f, 0.f, 0.f, 0.f, 0.f};
            acc[t] = z8;
            #pragma unroll
            for (int k = 0; k < 4; ++k) {
                acc[t] = __builtin_amdgcn_wmma_f32_16x16x4_f32(
                    /*neg_a=*/false, A[k], /*neg_b=*/false, Bf[t][k],
                    /*c_mod=*/(short)0, acc[t], /*reuse_a=*/false, /*reuse_b=*/false);
            }
        }

        // D: VGPR v, half h -> M = v + 8h, N = col. row_global = tile*16 + M.
        float* outp = out + (size_t)(tile * 16 + 8 * h) * IJ + col;
        #pragma unroll
        for (int t = 0; t < 4; ++t) {
            #pragma unroll
            for (int v = 0; v < 8; ++v) {
                outp[(size_t)v * IJ + t * 16] = acc[t][v];
            }
        }
    }
}

extern "C" void kernel_launch(void* const* d_in, const int* in_sizes, int n_in,
                              void* d_out, int out_size, void* d_ws, size_t ws_size,
                              hipStream_t stream) {
    const float* x     = (const float*)d_in[0];
    const float* a     = (const float*)d_in[1];
    const float* b     = (const float*)d_in[2];
    const float* c     = (const float*)d_in[3];
    const float* d     = (const float*)d_in[4];
    const float* alpha = (const float*)d_in[5];
    const float* beta  = (const float*)d_in[6];
    const float* gamma = (const float*)d_in[7];
    const float* delta = (const float*)d_in[8];
    const float* omega = (const float*)d_in[9];
    const float* Lam   = (const float*)d_in[10];
    float* out = (float*)d_out;

    const int tiles  = B_TOT / BPW;          // 16384
    const int blocks = tiles / WAVES;        // 2048
    nkat_wmma_kernel<<<dim3(blocks), dim3(THREADS), 0, stream>>>(
        x, a, b, c, d, alpha, beta, gamma, delta, omega, Lam, out);
}